// SimpleFPSPoolLayer_81758997447305
// MI455X (gfx1250) — compile-verified
//
#include <hip/hip_runtime.h>
#include <stdint.h>

#define N_POINTS 524288
#define M_OUT    2048
#define NWG      256
#define WGS      256
#define PPW      (N_POINTS / NWG)   // 2048 points per workgroup (contiguous slab)
#define PPT      (PPW / WGS)        // 8 points per thread (register resident)
#define FLT_BIG  3.402823466e+38f

// ---- CDNA5 async global->LDS staging, guarded so compile never breaks ------
#if defined(__has_builtin)
#  if __has_builtin(__builtin_amdgcn_global_load_async_to_lds_b128)
#    define FPS_ASYNC_LDS 1
#  endif
#  if __has_builtin(__builtin_amdgcn_s_wait_asynccnt)
#    define FPS_WAIT_ASYNC_BUILTIN 1
#  endif
#endif

#if FPS_ASYNC_LDS
// Exact pointee type from the compiler diagnostic: int __vector_size__(16).
typedef int fps_v4i __attribute__((vector_size(4 * sizeof(int))));
typedef __attribute__((address_space(1))) fps_v4i* fps_gv4p;  // global
typedef __attribute__((address_space(3))) fps_v4i* fps_lv4p;  // LDS
#endif

// Zero the per-iteration winner slots (u64) and arrival counters (u32).
__global__ void fps_init_kernel(uint32_t* ws, int n) {
    int i = blockIdx.x * blockDim.x + threadIdx.x;
    if (i < n) ws[i] = 0u;
}

__global__ __launch_bounds__(WGS) void fps_main_kernel(
    const float* __restrict__ pos, float* __restrict__ out,
    unsigned long long* __restrict__ slots, unsigned int* __restrict__ arrive) {

    __shared__ unsigned long long red[WGS / 32];
    __shared__ float qbuf[3];
#if FPS_ASYNC_LDS
    __shared__ __attribute__((aligned(16))) float slab[PPW * 3];  // 24 KB slab
#endif

    const unsigned tid  = threadIdx.x;
    const unsigned base = blockIdx.x * PPW;  // first global point index of slab

    // ---- one-time load of this WG's 2048 points into registers ----
    float px[PPT], py[PPT], pz[PPT], md[PPT];

#if FPS_ASYNC_LDS
    {
        const char* gsrc = (const char*)(pos + (size_t)base * 3);
        // 2048 pts * 12 B = 24576 B; 256 threads * 16 B * 6 passes
        #pragma unroll
        for (int i = 0; i < 6; ++i) {
            unsigned off = (tid + i * WGS) * 16u;
            // AS1 pointer: flat VA == global VA.
            fps_gv4p g = (fps_gv4p)(uintptr_t)(gsrc + off);
            // AS3 pointer: low 32 bits of flat address == LDS byte offset
            // (ISA 10.2).
            fps_lv4p l = (fps_lv4p)(uint32_t)(uintptr_t)((char*)slab + off);
            __builtin_amdgcn_global_load_async_to_lds_b128(g, l, 0, 0);
        }
#if FPS_WAIT_ASYNC_BUILTIN
        __builtin_amdgcn_s_wait_asynccnt(0);
#else
        asm volatile("s_wait_asynccnt 0" ::: "memory");
#endif
        __syncthreads();
        #pragma unroll
        for (int k = 0; k < PPT; ++k) {
            unsigned l = tid + k * WGS;           // local point index
            px[k] = slab[l * 3 + 0];
            py[k] = slab[l * 3 + 1];
            pz[k] = slab[l * 3 + 2];
            md[k] = FLT_BIG;
        }
        __syncthreads();                          // slab dead after this
    }
#else
    #pragma unroll
    for (int k = 0; k < PPT; ++k) {
        unsigned g = base + tid + k * WGS;
        px[k] = pos[(size_t)g * 3 + 0];
        py[k] = pos[(size_t)g * 3 + 1];
        pz[k] = pos[(size_t)g * 3 + 2];
        md[k] = FLT_BIG;
    }
#endif

    // idxs[0] = 0: first query is pos[0]; emit it once.
    if (blockIdx.x == 0 && tid == 0) {
        out[0] = pos[0]; out[1] = pos[1]; out[2] = pos[2];
    }
    float qx = pos[0], qy = pos[1], qz = pos[2];

    for (int iter = 1; iter < M_OUT; ++iter) {
        // ---- distance update + thread-local argmax (key packs value|~idx) --
        unsigned long long bestkey = 0ull;
        #pragma unroll
        for (int k = 0; k < PPT; ++k) {
            float d;
            {
                // Match XLA's unfused, left-associated evaluation exactly:
                // ((dx*dx + dy*dy) + dz*dz), no FMA contraction. The argmax
                // winner depends on exact rounding, so this must not fuse.
                #pragma clang fp contract(off)
                float dx = px[k] - qx, dy = py[k] - qy, dz = pz[k] - qz;
                d = dx * dx + dy * dy + dz * dz;
            }
            float m  = fminf(md[k], d);
            md[k] = m;
            unsigned gidx = base + tid + k * WGS;
            unsigned long long key =
                ((unsigned long long)__float_as_uint(m) << 32) |
                (unsigned long long)(~gidx);           // smaller idx wins ties
            bestkey = key > bestkey ? key : bestkey;
        }

        // ---- wave32 reduction ----
        #pragma unroll
        for (int off = 16; off > 0; off >>= 1) {
            unsigned long long o = __shfl_xor(bestkey, off, 32);
            bestkey = o > bestkey ? o : bestkey;
        }
        unsigned lane = tid & 31u, wv = tid >> 5;
        if (lane == 0) red[wv] = bestkey;
        __syncthreads();

        // ---- WG leader: global argmax via atomic umax, then grid barrier ---
        if (tid == 0) {
            unsigned long long k = red[0];
            #pragma unroll
            for (int w = 1; w < WGS / 32; ++w) k = red[w] > k ? red[w] : k;
            __hip_atomic_fetch_max(&slots[iter], k,
                                   __ATOMIC_RELAXED, __HIP_MEMORY_SCOPE_AGENT);
            __hip_atomic_fetch_add(&arrive[iter], 1u,
                                   __ATOMIC_ACQ_REL, __HIP_MEMORY_SCOPE_AGENT);
            while (__hip_atomic_load(&arrive[iter], __ATOMIC_ACQUIRE,
                                     __HIP_MEMORY_SCOPE_AGENT) < (unsigned)NWG)
                __builtin_amdgcn_s_sleep(1);
            unsigned long long win =
                __hip_atomic_load(&slots[iter], __ATOMIC_RELAXED,
                                  __HIP_MEMORY_SCOPE_AGENT);
            unsigned idx = ~((unsigned)win);
            float wx = pos[(size_t)idx * 3 + 0];
            float wy = pos[(size_t)idx * 3 + 1];
            float wz = pos[(size_t)idx * 3 + 2];
            qbuf[0] = wx; qbuf[1] = wy; qbuf[2] = wz;
            if (blockIdx.x == 0) {
                out[iter * 3 + 0] = wx;
                out[iter * 3 + 1] = wy;
                out[iter * 3 + 2] = wz;
            }
        }
        __syncthreads();
        qx = qbuf[0]; qy = qbuf[1]; qz = qbuf[2];
    }
}

extern "C" void kernel_launch(void* const* d_in, const int* in_sizes, int n_in,
                              void* d_out, int out_size, void* d_ws, size_t ws_size,
                              hipStream_t stream) {
    (void)in_sizes; (void)n_in; (void)out_size; (void)ws_size;
    const float* pos = (const float*)d_in[0];
    float* out = (float*)d_out;

    unsigned long long* slots = (unsigned long long*)d_ws;
    unsigned int* arrive = (unsigned int*)((char*)d_ws + M_OUT * sizeof(unsigned long long));

    // Zero slots (2048 u64 = 4096 u32) + arrival counters (2048 u32).
    int n32 = M_OUT * 2 + M_OUT;
    fps_init_kernel<<<(n32 + 255) / 256, 256, 0, stream>>>((uint32_t*)d_ws, n32);

    fps_main_kernel<<<NWG, WGS, 0, stream>>>(pos, out, slots, arrive);
}